// MultiBoxLoss_71047349010540
// MI455X (gfx1250) — compile-verified
//
#include <hip/hip_runtime.h>
#include <hip/hip_bf16.h>
#include <math.h>

#define NUM_CLASSES 81
#define BATCH 64
#define NPRI 8732
#define NOBJ 16
#define THR 0.5f

#define ROWSPW 16                 // rows per wave
#define WAVES 8                   // waves per block (256 threads)
#define ROWSPB (ROWSPW * WAVES)   // 128 rows per block
#define LDS_STRIDE 84             // 81 classes padded to 21*4 for WMMA K-chunks
#define SORTN 16384

typedef __attribute__((ext_vector_type(2))) float v2f;
typedef __attribute__((ext_vector_type(8))) float v8f;

__device__ __forceinline__ unsigned f32_sortkey(float f) {
  unsigned u = __float_as_uint(f);
  return (u & 0x80000000u) ? ~u : (u | 0x80000000u);
}

// ---------------------------------------------------------------- init
__global__ void init_kernel(float* sums) {
  if (threadIdx.x < 2) sums[threadIdx.x] = 0.0f;
}

// ---------------------------------------------------------------- match
// One block per batch. Reproduces match_one(): jaccard, per-prior best truth
// (first-index tie via strict >), per-truth best prior (first-index tie via
// packed u64 atomicMax), then the .at[].set overrides (last t wins).
__global__ __launch_bounds__(256) void match_kernel(
    const float* __restrict__ priors, const float* __restrict__ targets,
    int* __restrict__ btidx, int* __restrict__ conf_t,
    float* __restrict__ ovl, int* __restrict__ npos) {
  const int b = blockIdx.x;
  __shared__ float tr[NOBJ][4];
  __shared__ float lb[NOBJ];
  __shared__ unsigned long long bestp[NOBJ];
  __shared__ int cnt;
  const int tid = threadIdx.x;
  if (tid < NOBJ) {
    const float* tg = targets + ((size_t)b * NOBJ + tid) * 5;
    tr[tid][0] = tg[0]; tr[tid][1] = tg[1];
    tr[tid][2] = tg[2]; tr[tid][3] = tg[3];
    lb[tid] = tg[4];
    bestp[tid] = 0ull;
  }
  if (tid == 0) cnt = 0;
  __syncthreads();

  const size_t base = (size_t)b * NPRI;
  for (int p = tid; p < NPRI; p += blockDim.x) {
    float cx = 300.0f * priors[p * 4 + 0];
    float cy = 300.0f * priors[p * 4 + 1];
    float w  = 300.0f * priors[p * 4 + 2];
    float h  = 300.0f * priors[p * 4 + 3];
    float px1 = cx - w * 0.5f, py1 = cy - h * 0.5f;
    float px2 = cx + w * 0.5f, py2 = cy + h * 0.5f;
    float areab = (px2 - px1) * (py2 - py1);
    float bestv = -1.0f; int besti = 0;
    for (int t = 0; t < NOBJ; ++t) {
      float ix1 = fmaxf(tr[t][0], px1), iy1 = fmaxf(tr[t][1], py1);
      float ix2 = fminf(tr[t][2], px2), iy2 = fminf(tr[t][3], py2);
      float iw = fmaxf(ix2 - ix1, 0.0f), ih = fmaxf(iy2 - iy1, 0.0f);
      float inter = iw * ih;
      float areaa = (tr[t][2] - tr[t][0]) * (tr[t][3] - tr[t][1]);
      float o = inter / (areaa + areab - inter);
      if (o > bestv) { bestv = o; besti = t; }
      unsigned long long key =
          ((unsigned long long)f32_sortkey(o) << 32) |
          (unsigned long long)(0xFFFFFFFFu - (unsigned)p);
      atomicMax(&bestp[t], key);
    }
    btidx[base + p] = besti;
    ovl[base + p] = bestv;
  }
  __syncthreads();
  if (tid == 0) {  // serialized so duplicate best-prior gets last-t (set semantics)
    for (int t = 0; t < NOBJ; ++t) {
      unsigned low = (unsigned)(bestp[t] & 0xFFFFFFFFull);
      int bp = (int)(0xFFFFFFFFu - low);
      ovl[base + bp] = 2.0f;
      btidx[base + bp] = t;
    }
  }
  __syncthreads();
  int local = 0;
  for (int p = tid; p < NPRI; p += blockDim.x) {
    float o = ovl[base + p];
    int bt = btidx[base + p];
    int c = (o < THR) ? 0 : ((int)lb[bt] + 1);
    conf_t[base + p] = c;
    if (c > 0) local++;
  }
  atomicAdd(&cnt, local);
  __syncthreads();
  if (tid == 0) npos[b] = cnt;
}

// ---------------------------------------------------------------- conf/loc loss
// Streams conf_data through LDS with async-to-LDS copies; logsumexp row sums
// via chained V_WMMA_F32_16X16X4_F32 (A = exp tiles, B = ones).
__global__ __launch_bounds__(256) void conf_loss_kernel(
    const float* __restrict__ loc_data, const float* __restrict__ conf_data,
    const float* __restrict__ priors, const float* __restrict__ targets,
    const int* __restrict__ btidx, const int* __restrict__ conf_t,
    float* __restrict__ lca, float* __restrict__ sums) {
  __shared__ float smem[WAVES * ROWSPW * LDS_STRIDE];
  const int b = blockIdx.x;
  const int wave = threadIdx.x >> 5;
  const int lane = threadIdx.x & 31;
  const int r0 = blockIdx.y * ROWSPB + wave * ROWSPW;
  float lloss = 0.0f;

  if (r0 < NPRI) {
    const int rows = min(ROWSPW, NPRI - r0);
    const int ndw = rows * NUM_CLASSES;
    const float* gbase = conf_data + ((size_t)b * NPRI + r0) * NUM_CLASSES;
    // low 32 bits of the generic address of a __shared__ pointer == LDS offset
    const unsigned wbase_b =
        (unsigned)(uintptr_t)(&smem[wave * ROWSPW * LDS_STRIDE]);
    for (int i = lane; i < ndw; i += 32) {
      int row = i / NUM_CLASSES;
      int col = i - row * NUM_CLASSES;
      unsigned laddr = wbase_b + (unsigned)(row * LDS_STRIDE + col) * 4u;
      unsigned long long ga = (unsigned long long)(uintptr_t)(gbase + i);
      asm volatile("global_load_async_to_lds_b32 %0, %1, off"
                   :: "v"(laddr), "v"(ga) : "memory");
    }
    asm volatile("s_wait_asynccnt 0" ::: "memory");

    // WMMA A-layout: lane holds row (lane&15); lane-halves split K pairs.
    const float* rowp =
        &smem[wave * ROWSPW * LDS_STRIDE + (lane & 15) * LDS_STRIDE];
    const int khalf = (lane >> 4) << 1;  // 0 or 2

    float mx = -INFINITY;
#pragma unroll
    for (int c = 0; c < 21; ++c) {
      int kb = 4 * c + khalf;
      float x0 = rowp[kb], x1 = rowp[kb + 1];
      if (kb < NUM_CLASSES) mx = fmaxf(mx, x0);
      if (kb + 1 < NUM_CLASSES) mx = fmaxf(mx, x1);
    }
    mx = fmaxf(mx, __shfl_xor(mx, 16, 32));  // full-row max in both halves

    v8f acc = {};
    v2f bones; bones.x = 1.0f; bones.y = 1.0f;  // B = ones(4x16): D=row sums
#pragma unroll
    for (int c = 0; c < 21; ++c) {
      int kb = 4 * c + khalf;
      float x0 = rowp[kb], x1 = rowp[kb + 1];
      v2f a;
      a.x = (kb < NUM_CLASSES) ? expf(x0 - mx) : 0.0f;
      a.y = (kb + 1 < NUM_CLASSES) ? expf(x1 - mx) : 0.0f;
      acc = __builtin_amdgcn_wmma_f32_16x16x4_f32(
          false, a, false, bones, (short)0, acc, false, false);
    }
    // D layout: lane0 vgpr r = rows 0..7, lane16 vgpr r = rows 8..15
    float rs = 0.0f;
#pragma unroll
    for (int r = 0; r < 8; ++r) {
      float s0 = __shfl(acc[r], 0, 32);
      float s1 = __shfl(acc[r], 16, 32);
      if (lane == r)     rs = s0;
      if (lane == r + 8) rs = s1;
    }

    const int row = r0 + lane;
    if (lane < 16 && row < NPRI) {
      const size_t gi = (size_t)b * NPRI + row;
      int ct = conf_t[gi];
      float g = rowp[ct];
      float lse = mx + logf(rs);
      lca[gi] = lse - g;
      if (ct > 0) {  // smooth-L1 for positives
        int bt = btidx[gi];
        const float* tg = targets + ((size_t)b * NOBJ + bt) * 5;
        float tx1 = tg[0], ty1 = tg[1], tx2 = tg[2], ty2 = tg[3];
        float pcx = 300.0f * priors[row * 4 + 0];
        float pcy = 300.0f * priors[row * 4 + 1];
        float pw  = 300.0f * priors[row * 4 + 2];
        float ph  = 300.0f * priors[row * 4 + 3];
        float lt0 = ((tx1 + tx2) * 0.5f - pcx) / (0.1f * pw);
        float lt1 = ((ty1 + ty2) * 0.5f - pcy) / (0.1f * ph);
        float lt2 = logf((tx2 - tx1) / pw) / 0.2f;
        float lt3 = logf((ty2 - ty1) / ph) / 0.2f;
        const float* lp = loc_data + gi * 4;
        float d[4] = {lp[0] - lt0, lp[1] - lt1, lp[2] - lt2, lp[3] - lt3};
        float s = 0.0f;
#pragma unroll
        for (int q = 0; q < 4; ++q) {
          float ad = fabsf(d[q]);
          s += (ad < 1.0f) ? 0.5f * d[q] * d[q] : ad - 0.5f;
        }
        lloss = s;
      }
    }
  }
  for (int off = 16; off > 0; off >>= 1) lloss += __shfl_down(lloss, off, 32);
  if (lane == 0 && lloss != 0.0f) atomicAdd(&sums[0], lloss);
}

// ---------------------------------------------------------------- mining
// One 1024-thread block per batch; 128 KB LDS bitonic sort (CDNA5 320KB WGP).
// Key = (~sortable(mined) << 32) | index  -> ascending == stable desc argsort.
__global__ __launch_bounds__(1024) void mine_kernel(
    const float* __restrict__ lca, const int* __restrict__ conf_t,
    const int* __restrict__ npos, float* __restrict__ sums) {
  __shared__ unsigned long long keys[SORTN];
  __shared__ float part[1024];
  const int b = blockIdx.x;
  const int tid = threadIdx.x;
  const size_t base = (size_t)b * NPRI;

  for (int i = tid; i < SORTN; i += 1024) {
    unsigned long long k;
    if (i < NPRI) {
      int ct = conf_t[base + i];
      float mined = (ct > 0) ? 0.0f : lca[base + i];
      unsigned hi = ~f32_sortkey(mined);
      k = ((unsigned long long)hi << 32) | (unsigned)i;
    } else {
      k = 0xFFFFFFFFFFFFFFFFull;  // sorts to the end
    }
    keys[i] = k;
  }
  __syncthreads();
  for (int k = 2; k <= SORTN; k <<= 1) {
    for (int j = k >> 1; j > 0; j >>= 1) {
      for (int i = tid; i < SORTN; i += 1024) {
        int ixj = i ^ j;
        if (ixj > i) {
          unsigned long long a = keys[i], c2 = keys[ixj];
          bool up = ((i & k) == 0);
          if ((a > c2) == up) { keys[i] = c2; keys[ixj] = a; }
        }
      }
      __syncthreads();
    }
  }
  // loss_c = sum over pos  +  sum over (top-num_neg of mined) excluding pos
  float local = 0.0f;
  for (int i = tid; i < NPRI; i += 1024)
    if (conf_t[base + i] > 0) local += lca[base + i];
  int nn = min(3 * npos[b], NPRI - 1);
  for (int i = tid; i < nn; i += 1024) {
    int idx = (int)(keys[i] & 0xFFFFFFFFull);
    if (conf_t[base + idx] == 0) local += lca[base + idx];
  }
  part[tid] = local;
  __syncthreads();
  for (int s = 512; s > 0; s >>= 1) {
    if (tid < s) part[tid] += part[tid + s];
    __syncthreads();
  }
  if (tid == 0) atomicAdd(&sums[1], part[0]);
}

// ---------------------------------------------------------------- finalize
__global__ void finalize_kernel(const float* __restrict__ sums,
                                const int* __restrict__ npos,
                                float* __restrict__ out) {
  int i = threadIdx.x;
  if (i < BATCH) {
    float n = (float)npos[i];
    out[i] = sums[0] / n;
    out[BATCH + i] = sums[1] / n;
  }
}

// ---------------------------------------------------------------- launch
extern "C" void kernel_launch(void* const* d_in, const int* in_sizes, int n_in,
                              void* d_out, int out_size, void* d_ws,
                              size_t ws_size, hipStream_t stream) {
  (void)in_sizes; (void)n_in; (void)out_size; (void)ws_size;
  const float* loc     = (const float*)d_in[0];
  const float* conf    = (const float*)d_in[1];
  const float* priors  = (const float*)d_in[2];
  const float* targets = (const float*)d_in[3];
  float* out = (float*)d_out;

  const size_t BP = (size_t)BATCH * NPRI;
  int*   btidx  = (int*)d_ws;
  int*   conf_t = btidx + BP;
  float* ovl    = (float*)(conf_t + BP);
  float* lca    = ovl + BP;
  int*   npos   = (int*)(lca + BP);
  float* sums   = (float*)(npos + BATCH);

  init_kernel<<<1, 64, 0, stream>>>(sums);
  match_kernel<<<BATCH, 256, 0, stream>>>(priors, targets, btidx, conf_t, ovl,
                                          npos);
  dim3 grid2(BATCH, (NPRI + ROWSPB - 1) / ROWSPB);
  conf_loss_kernel<<<grid2, 256, 0, stream>>>(loc, conf, priors, targets,
                                              btidx, conf_t, lca, sums);
  mine_kernel<<<BATCH, 1024, 0, stream>>>(lca, conf_t, npos, sums);
  finalize_kernel<<<1, 64, 0, stream>>>(sums, npos, out);
}